// MixITLoss_12025908429479
// MI455X (gfx1250) — compile-verified
//
#include <hip/hip_runtime.h>
#include <math.h>

// ---------------------------------------------------------------------------
// MixIT loss for MI455X (gfx1250).
//
// Everything reduces to per-batch 6x6 Gram matrices of X = [est0..est3,m1,m2]:
//   G[i][j] = sum_t X[i][t]*X[j][t]
// computed with V_WMMA_F32_16X16X4_F32 (fp32 matrix op -> exact-precision
// path), then a tiny 16-combo epilogue. Bandwidth-bound: ~49MB @ 23.3TB/s.
// ---------------------------------------------------------------------------

typedef __attribute__((ext_vector_type(2))) float v2f;
typedef __attribute__((ext_vector_type(8))) float v8f;

#define NB     32          // batch
#define NS     4           // estimated sources
#define TLEN   64000       // time samples
#define NROW   6           // rows of X: 4 sources + m1 + m2
#define NCHUNK 32          // T-chunks per batch (TLEN/NCHUNK = 2000, %4 == 0)
#define NGRAM  36          // 6x6 Gram entries stored per partial
#define NCOMB  16          // 2^4 assignment combinations

// ---------------------------------------------------------------------------
// Kernel 1: one wave (32 lanes) per (batch, chunk). Each iteration feeds a
// 16x4 fp32 tile (rows of X x 4 timesteps) to WMMA as both A and B, so the
// f32 accumulator collects D[m][n] += sum_k X[m][t+k]*X[n][t+k].
//
// A-layout (16x4 f32): lane l<16 -> M=l, VGPR0/1 = K0/K1; lanes 16-31 = K2/K3.
// B-layout (4x16 f32): VGPR0 = rows {K0 | K2} striped over lane halves, VGPR1
// = {K1 | K3}; for B = A^T this is bit-identical to the A operand, so we pass
// the same float2 register twice. Rows 6..15 alias row 0 -> their products
// land in Gram entries we never read; no masking, no divergence.
// ---------------------------------------------------------------------------
__global__ __launch_bounds__(32)
void mixit_gram_wmma(const float* __restrict__ est,
                     const float* __restrict__ m1,
                     const float* __restrict__ m2,
                     float* __restrict__ partial)
{
    const int blk   = blockIdx.x;        // b * NCHUNK + chunk
    const int b     = blk / NCHUNK;
    const int chunk = blk % NCHUNK;
    const int lane  = threadIdx.x;       // 0..31 (wave32)
    const int row   = lane & 15;         // matrix row this lane feeds
    const int hi    = lane >> 4;         // 0: K=0,1   1: K=2,3

    const int tc = TLEN / NCHUNK;        // 2000, multiple of 4
    const int t0 = chunk * tc;

    // Row pointer: 0..3 -> sources, 4 -> m1, 5 -> m2, 6..15 -> alias row 0.
    const int r = (row < NROW) ? row : 0;
    const float* rp;
    if (r < NS)       rp = est + ((size_t)b * NS + r) * TLEN;
    else if (r == 4)  rp = m1 + (size_t)b * TLEN;
    else              rp = m2 + (size_t)b * TLEN;

    const float* p = rp + t0 + 2 * hi;

    v8f acc = {};
    #pragma unroll 4
    for (int i = 0; i < tc; i += 4) {
        v2f a;
        a.x = p[0];
        a.y = p[1];
        // D = A * A^T + D  (same register as A and B operand)
        acc = __builtin_amdgcn_wmma_f32_16x16x4_f32(
            /*neg_a=*/false, a, /*neg_b=*/false, a,
            /*c_mod=*/(short)0, acc, /*reuse_a=*/false, /*reuse_b=*/false);
        p += 4;
    }

    // C/D layout: VGPR m, lanes 0..15 hold D[M=m][N=lane] (m<8).
    // Lanes 0..5 own columns 0..5 of the 6x6 Gram -> spill 36 floats.
    if (lane < NROW) {
        float* out = partial + (size_t)blk * NGRAM + lane;
        out[0]  = acc[0];
        out[6]  = acc[1];
        out[12] = acc[2];
        out[18] = acc[3];
        out[24] = acc[4];
        out[30] = acc[5];
    }
}

// ---------------------------------------------------------------------------
// Kernel 2: single block. Reduce chunk partials -> per-batch Gram, evaluate
// all 16 combos, sum over batch, argmin/min.
// ---------------------------------------------------------------------------
__global__ __launch_bounds__(256)
void mixit_finalize(const float* __restrict__ partial,
                    float* __restrict__ out, int out_size)
{
    __shared__ float sG[NB][NGRAM];      // per-batch Gram
    __shared__ float sLoss[NCOMB][NB];   // per (combo, batch) loss
    __shared__ float sK[NCOMB];          // per-combo total loss
    const int tid = threadIdx.x;

    // Stage A: sum partials over chunks (all traffic is L2-resident, tiny).
    for (int idx = tid; idx < NB * NGRAM; idx += blockDim.x) {
        const int b = idx / NGRAM, j = idx % NGRAM;
        const float* p = partial + ((size_t)b * NCHUNK) * NGRAM + j;
        float s = 0.0f;
        for (int ch = 0; ch < NCHUNK; ++ch) s += p[(size_t)ch * NGRAM];
        sG[b][j] = s;
    }
    __syncthreads();

    // Stage B: loss for every (combo k, batch b).
    for (int idx = tid; idx < NCOMB * NB; idx += blockDim.x) {
        const int k = idx / NB, b = idx % NB;
        const float* G = &sG[b][0];

        float mk[NS], ck[NS];
        #pragma unroll
        for (int s = 0; s < NS; ++s) {
            // itertools.product([0,1], repeat=4): first source = MSB of k
            mk[s] = (float)((k >> (3 - s)) & 1);
            ck[s] = 1.0f - mk[s];
        }

        float b0 = 0.0f, b1 = 0.0f, d0 = 0.0f, d1 = 0.0f;
        #pragma unroll
        for (int s = 0; s < NS; ++s) {
            #pragma unroll
            for (int s2 = 0; s2 < NS; ++s2) {
                const float g = G[s * NROW + s2];
                b0 += mk[s] * mk[s2] * g;   // || mix0 ||^2
                b1 += ck[s] * ck[s2] * g;   // || mix1 ||^2
            }
            d0 += mk[s] * G[s * NROW + 4];  // <mix0, m1>
            d1 += ck[s] * G[s * NROW + 5];  // <mix1, m2>
        }
        const float M1 = G[4 * NROW + 4];   // ||m1||^2
        const float M2 = G[5 * NROW + 5];   // ||m2||^2
        const float a0 = b0 - 2.0f * d0 + M1;  // ||mix0 - m1||^2
        const float a1 = b1 - 2.0f * d1 + M2;  // ||mix1 - m2||^2

        const float loss =
            10.0f * (__log10f(a0 + 30.0f * b0) - __log10f(b0)) +
            10.0f * (__log10f(a1 + 30.0f * b1) - __log10f(b1));
        sLoss[k][b] = loss;
    }
    __syncthreads();

    // Stage C: sum over batch per combo.
    if (tid < NCOMB) {
        float s = 0.0f;
        for (int b = 0; b < NB; ++b) s += sLoss[tid][b];
        sK[tid] = s;
    }
    __syncthreads();

    // Stage D: first-occurrence argmin + min (matches jnp.argmin/min).
    if (tid == 0) {
        int   bi = 0;
        float bv = sK[0];
        for (int k = 1; k < NCOMB; ++k) {
            if (sK[k] < bv) { bv = sK[k]; bi = k; }
        }
        out[0] = (float)bi;
        if (out_size > 1) out[1] = bv;
    }
}

// ---------------------------------------------------------------------------
extern "C" void kernel_launch(void* const* d_in, const int* in_sizes, int n_in,
                              void* d_out, int out_size, void* d_ws, size_t ws_size,
                              hipStream_t stream) {
    (void)in_sizes; (void)n_in; (void)ws_size;
    const float* est = (const float*)d_in[0];   // [32, 4, 64000]
    const float* m1  = (const float*)d_in[1];   // [32, 64000]
    const float* m2  = (const float*)d_in[2];   // [32, 64000]
    float* partial = (float*)d_ws;              // [NB*NCHUNK, 36] = 144 KB

    mixit_gram_wmma<<<dim3(NB * NCHUNK), dim3(32), 0, stream>>>(est, m1, m2, partial);
    mixit_finalize<<<dim3(1), dim3(256), 0, stream>>>(partial, (float*)d_out, out_size);
}